// Gram_83743272337809
// MI455X (gfx1250) — compile-verified
//
#include <hip/hip_runtime.h>
#include <math.h>

// g[b,i,j] = cos(phi_i + phi_j) = c_i*c_j - s_i*s_j with c = minmax-scaled/clipped x,
// s = sqrt(1-c^2).  Rank-2 outer product per batch => V_WMMA_F32_16X16X4_F32 tiles.
// Pure store-bandwidth problem: 512 MiB out @ 23.3 TB/s ~ 23 us floor.
// Output is write-once and 2.7x the 192MB L2 -> non-temporal stores.

typedef __attribute__((ext_vector_type(2))) float v2f;
typedef __attribute__((ext_vector_type(8))) float v8f;

constexpr int kN     = 512;        // hard-coded N in reference
constexpr int kTiles = kN / 16;    // 32 tiles per dimension
constexpr float kEps = 1e-6f;

__global__ __launch_bounds__(256) void gram_wmma_kernel(const float* __restrict__ x,
                                                        float* __restrict__ out) {
    const int b   = blockIdx.x;
    const int tid = threadIdx.x;

    __shared__ float sC[kN];
    __shared__ float sS[kN];
    __shared__ float rmn[256];
    __shared__ float rmx[256];

    // ---- Phase A: load row, min/max reduce, build c and s in LDS ----
    const float x0 = x[(size_t)b * kN + tid];
    const float x1 = x[(size_t)b * kN + tid + 256];
    rmn[tid] = fminf(x0, x1);
    rmx[tid] = fmaxf(x0, x1);
    __syncthreads();
#pragma unroll
    for (int s = 128; s > 0; s >>= 1) {
        if (tid < s) {
            rmn[tid] = fminf(rmn[tid], rmn[tid + s]);
            rmx[tid] = fmaxf(rmx[tid], rmx[tid + s]);
        }
        __syncthreads();
    }
    const float mn  = rmn[0];
    const float mx  = rmx[0];
    const float inv = 1.0f / (mx - mn);
    const float off = -(mx + mn);

    float c0 = (2.0f * x0 + off) * inv;
    float c1 = (2.0f * x1 + off) * inv;
    c0 = fminf(fmaxf(c0, -1.0f + kEps), 1.0f - kEps);
    c1 = fminf(fmaxf(c1, -1.0f + kEps), 1.0f - kEps);
    sC[tid]       = c0;
    sC[tid + 256] = c1;
    sS[tid]       = sqrtf(fmaxf(1.0f - c0 * c0, 0.0f));
    sS[tid + 256] = sqrtf(fmaxf(1.0f - c1 * c1, 0.0f));
    __syncthreads();

    // ---- Phase B: 16x16 output tiles via V_WMMA_F32_16X16X4_F32 ----
    const int   wave   = tid >> 5;
    const int   lane   = tid & 31;
    const int   l16    = lane & 15;
    const float msk    = (lane < 16) ? 1.0f : 0.0f;   // lanes 16-31 carry K=2,3 == 0
    const int   rowadd = (lane < 16) ? 0 : 8;         // D: upper half-wave holds rows v+8

    float* outb = out + (size_t)b * kN * kN;

    for (int ti = wave; ti < kTiles; ti += 8) {
        // A fragment (16x4, rows ti*16..+15): VGPR0=K0=c, VGPR1=K1=s (lanes 0-15)
        v2f a;
        a.x = sC[ti * 16 + l16] * msk;
        a.y = sS[ti * 16 + l16] * msk;
        const int row0 = ti * 16 + rowadd;

        for (int tj = 0; tj < kTiles; ++tj) {
            // B fragment (4x16, cols tj*16..+15): row K0=c, row K1=-s (lanes 0-15)
            v2f bf;
            bf.x =  sC[tj * 16 + l16] * msk;
            bf.y = -sS[tj * 16 + l16] * msk;

            v8f acc = {};
            // (neg_a, A, neg_b, B, c_mod, C, reuse_a, reuse_b)
            v8f d = __builtin_amdgcn_wmma_f32_16x16x4_f32(
                false, a, false, bf, (short)0, acc, false, false);

            float* p = outb + (size_t)row0 * kN + (tj * 16 + l16);
#pragma unroll
            for (int v = 0; v < 8; ++v) {
                __builtin_nontemporal_store(d[v], p + (size_t)v * kN);
            }
        }
    }
}

extern "C" void kernel_launch(void* const* d_in, const int* in_sizes, int n_in,
                              void* d_out, int out_size, void* d_ws, size_t ws_size,
                              hipStream_t stream) {
    const float* x   = (const float*)d_in[0];
    float*       out = (float*)d_out;
    const int    B   = in_sizes[0] / kN;   // 512 batches
    gram_wmma_kernel<<<B, 256, 0, stream>>>(x, out);
}